// LaplacianOperator_33303176413246
// MI455X (gfx1250) — compile-verified
//
#include <hip/hip_runtime.h>
#include <stdint.h>

// Sparse COO Laplacian apply: out[row[e], 0:8] += vals[e] * field[col[e], 0:8]
//
// Roofline: 288 MFLOP vs ~424 MB mandatory traffic -> memory bound (~18us floor
// at 23.3 TB/s). field (32MB) + out (32MB) are kept L2-resident (192MB L2);
// the 360MB of COO streams are read once with non-temporal hints so they do
// not evict the resident set. Scatter uses non-returning global_atomic_add_f32.

typedef __attribute__((ext_vector_type(4))) float v4f;

#define BLOCK 256
#define NNZ_PER_THREAD 4

// Emitted first so the disasm snippet shows this kernel's instruction mix.
__global__ __launch_bounds__(BLOCK) void lap_coo_spmm_kernel(
    const float* __restrict__ vals,
    const int64_t* __restrict__ rows,
    const int64_t* __restrict__ cols,
    const float* __restrict__ field,
    float* __restrict__ out,
    long long nnz,
    long long stride) {  // stride = gridDim.x * BLOCK
  long long base = (long long)blockIdx.x * BLOCK + threadIdx.x;

#pragma unroll
  for (int it = 0; it < NNZ_PER_THREAD; ++it) {
    long long e = base + (long long)it * stride;
    if (e < nnz) {
      // Pull the next grid-stride tile of the three streams toward L2.
      // Masked so ~2 lanes per wave issue it: one prefetch covers a line.
      long long pe = e + stride;
      if (((threadIdx.x & 15u) == 0u) && (pe < nnz)) {
        __builtin_prefetch(vals + pe, 0, 0);  // global_prefetch_b8
        __builtin_prefetch(rows + pe, 0, 0);
        __builtin_prefetch(cols + pe, 0, 0);
      }

      // Streaming reads: read-once, non-temporal (th:TH_LOAD_NT) so the
      // 360MB of COO data does not evict field/out from L2.
      float   v   = __builtin_nontemporal_load(vals + e);
      int64_t r64 = __builtin_nontemporal_load(rows + e);
      int64_t c64 = __builtin_nontemporal_load(cols + e);
      // Indices are < 1M: truncate to 32-bit to halve address-math VALU work.
      int r = (int)r64;
      int c = (int)c64;

      // Gather one 32B field row (L2-resident, regular temporal policy).
      const v4f* frow = (const v4f*)(field + ((long long)c << 3));
      v4f f0 = frow[0];
      v4f f1 = frow[1];

      // Scatter-add 8 floats: non-returning global_atomic_add_f32, all 8
      // in flight simultaneously (STOREcnt tracked, no data dependence).
      float* orow = out + ((long long)r << 3);
      unsafeAtomicAdd(orow + 0, v * f0.x);
      unsafeAtomicAdd(orow + 1, v * f0.y);
      unsafeAtomicAdd(orow + 2, v * f0.z);
      unsafeAtomicAdd(orow + 3, v * f0.w);
      unsafeAtomicAdd(orow + 4, v * f1.x);
      unsafeAtomicAdd(orow + 5, v * f1.y);
      unsafeAtomicAdd(orow + 6, v * f1.z);
      unsafeAtomicAdd(orow + 7, v * f1.w);
    }
  }
}

__global__ __launch_bounds__(BLOCK) void lap_zero_kernel(float* __restrict__ out,
                                                         int n4) {
  int i = blockIdx.x * BLOCK + threadIdx.x;
  if (i < n4) {
    ((v4f*)out)[i] = (v4f){0.0f, 0.0f, 0.0f, 0.0f};
  }
}

extern "C" void kernel_launch(void* const* d_in, const int* in_sizes, int n_in,
                              void* d_out, int out_size, void* d_ws, size_t ws_size,
                              hipStream_t stream) {
  // setup_inputs() order: field[n*8] f32, lap_vals[nnz] f32,
  //                       lap_row[nnz] i64, lap_col[nnz] i64, n_vertices scalar
  const float*   field = (const float*)d_in[0];
  const float*   vals  = (const float*)d_in[1];
  const int64_t* rows  = (const int64_t*)d_in[2];
  const int64_t* cols  = (const int64_t*)d_in[3];
  float* out = (float*)d_out;

  long long nnz = (long long)in_sizes[1];

  // 1) Zero the output (harness poisons it; atomics accumulate into zeros).
  int n4 = out_size / 4;  // out_size floats, 4 per v4f store
  int zgrid = (n4 + BLOCK - 1) / BLOCK;
  lap_zero_kernel<<<zgrid, BLOCK, 0, stream>>>(out, n4);

  // 2) Gather-multiply-scatter over all nonzeros, 4 per thread, grid-stride.
  long long elems_per_block = (long long)BLOCK * NNZ_PER_THREAD;
  int grid = (int)((nnz + elems_per_block - 1) / elems_per_block);
  long long stride = (long long)grid * BLOCK;
  lap_coo_spmm_kernel<<<grid, BLOCK, 0, stream>>>(vals, rows, cols, field, out,
                                                  nnz, stride);
}